// FlashAttentionPipelined_57784490000872
// MI455X (gfx1250) — compile-verified
//
#include <hip/hip_runtime.h>
#include <stdint.h>

typedef __attribute__((ext_vector_type(2))) float v2f;
typedef __attribute__((ext_vector_type(4))) float v4f;
typedef __attribute__((ext_vector_type(8))) float v8f;

#define D_DIM   128
#define N_SEQ   4096
#define BR      128
#define BC      64
#define NWAVE   8
#define THREADS 256
#define QSTR    132   // padded row stride (floats) -> bank-conflict-free WMMA A/B fetch
#define KSTR    132
#define VSTR    68
#define PSTR    68

// LDS layout (floats)
#define SM_QS     0
#define SM_KS0    (SM_QS  + BR * QSTR)
#define SM_KS1    (SM_KS0 + BC * KSTR)
#define SM_VT0    (SM_KS1 + BC * KSTR)
#define SM_VT1    (SM_VT0 + D_DIM * VSTR)
#define SM_PS     (SM_VT1 + D_DIM * VSTR)
#define SM_TOTAL  (SM_PS  + NWAVE * 16 * PSTR)   // 59904 floats = 239616 B

__device__ __forceinline__ v8f wmma4(v2f a, v2f b, v8f c) {
  // fp32 WMMA: D(16x16) = A(16x4) * B(4x16) + C, full fp32 accumulate
  return __builtin_amdgcn_wmma_f32_16x16x4_f32(false, a, false, b, (short)0, c,
                                               false, false);
}

__device__ __forceinline__ void async_b128(uint32_t lds_off, const float* g) {
  // CDNA5 async global -> LDS copy (tracked by ASYNCcnt)
  asm volatile("global_load_async_to_lds_b128 %0, %1, off"
               :: "v"(lds_off), "v"(g) : "memory");
}

__device__ __forceinline__ void wait_async0() {
  asm volatile("s_wait_asynccnt 0" ::: "memory");
}

// Copy ROWS x 128 fp32 row-major tile from global to LDS (padded stride STR)
template <int ROWS, int STR>
__device__ __forceinline__ void async_tile(const float* lds_base, const float* g,
                                           int tid) {
  uint32_t l0 = (uint32_t)(uintptr_t)lds_base;
  constexpr int CHUNKS = ROWS * (D_DIM / 4);
#pragma unroll
  for (int c0 = 0; c0 < CHUNKS; c0 += THREADS) {
    int c = c0 + tid;
    int r = c >> 5;
    int col = (c & 31) << 2;
    async_b128(l0 + (uint32_t)(r * STR + col) * 4u, g + (size_t)r * D_DIM + col);
  }
}

__global__ __launch_bounds__(THREADS, 1)
void fa_fwd_f32_wmma(const float* __restrict__ Q, const float* __restrict__ K,
                     const float* __restrict__ V, float* __restrict__ O) {
  extern __shared__ float smem[];
  float* Qs  = smem + SM_QS;
  float* Ks0 = smem + SM_KS0;
  float* Ks1 = smem + SM_KS1;
  float* Vt0 = smem + SM_VT0;
  float* Vt1 = smem + SM_VT1;
  float* Ps  = smem + SM_PS;

  const int tid  = threadIdx.x;
  const int lane = tid & 31;
  const int wv   = tid >> 5;
  const int hi   = lane >> 4;   // half-wave select (A/B K-pair select)
  const int ln   = lane & 15;   // M (A) / N (B) index within 16

  const size_t head = blockIdx.y;                       // b*H + h
  const float* Qh = Q + head * (size_t)(N_SEQ * D_DIM);
  const float* Kh = K + head * (size_t)(N_SEQ * D_DIM);
  const float* Vh = V + head * (size_t)(N_SEQ * D_DIM);
  float*       Oh = O + head * (size_t)(N_SEQ * D_DIM);
  const int q0 = blockIdx.x * BR;

  // ---- prologue: async Q + K(0), V(0) prefetched to VGPRs ----
  async_tile<BR, QSTR>(Qs, Qh + (size_t)q0 * D_DIM, tid);
  async_tile<BC, KSTR>(Ks0, Kh, tid);

  v4f vreg[8];
#pragma unroll
  for (int i = 0; i < 8; ++i) {
    int c = tid + i * THREADS;
    int r = c >> 5, col = (c & 31) << 2;
    vreg[i] = *(const v4f*)(Vh + (size_t)r * D_DIM + col);
  }

  v8f Oacc[8];
  float mrow[8], lrow[8];
#pragma unroll
  for (int ot = 0; ot < 8; ++ot)
#pragma unroll
    for (int v = 0; v < 8; ++v) Oacc[ot][v] = 0.0f;
#pragma unroll
  for (int v = 0; v < 8; ++v) { mrow[v] = -1e30f; lrow[v] = 0.0f; }

  float* Pw = Ps + wv * 16 * PSTR;            // this wave's P staging tile
  const float SCL = 0.08838834764831845f * 1.44269504088896340f; // 1/sqrt(d)*log2e

  const int TC = N_SEQ / BC;
  for (int j = 0; j < TC; ++j) {
    float* Kb = (j & 1) ? Ks1 : Ks0;
    float* Vb = (j & 1) ? Vt1 : Vt0;

    wait_async0();                            // K(j) (and Q on j==0) landed in LDS

    // transpose-store V(j) from VGPRs: Vt[n][k] = V[k][n]
#pragma unroll
    for (int i = 0; i < 8; ++i) {
      int c = tid + i * THREADS;
      int r = c >> 5, col = (c & 31) << 2;
#pragma unroll
      for (int ii = 0; ii < 4; ++ii) Vb[(col + ii) * VSTR + r] = vreg[i][ii];
    }

    if (j + 1 < TC) {                         // prefetch next tile (overlaps compute)
      float* Kn = (j & 1) ? Ks0 : Ks1;
      async_tile<BC, KSTR>(Kn, Kh + (size_t)(j + 1) * BC * D_DIM, tid);
      const float* Vn = Vh + (size_t)(j + 1) * BC * D_DIM;
#pragma unroll
      for (int i = 0; i < 8; ++i) {
        int c = tid + i * THREADS;
        int r = c >> 5, col = (c & 31) << 2;
        vreg[i] = *(const v4f*)(Vn + (size_t)r * D_DIM + col);
      }
    }
    __syncthreads();                          // K(j)/V(j) visible to all waves

    // ---- S = Q * K^T  (fp32 WMMA, K-dim = 128 in steps of 4) ----
    v8f S[4];
#pragma unroll
    for (int nt = 0; nt < 4; ++nt)
#pragma unroll
      for (int v = 0; v < 8; ++v) S[nt][v] = 0.0f;

    const float* Qa = Qs + (wv * 16 + ln) * QSTR + 2 * hi;
#pragma unroll
    for (int kk = 0; kk < 32; ++kk) {
      v2f a = *(const v2f*)(Qa + kk * 4);     // A: Q[m][k..k+1]
#pragma unroll
      for (int nt = 0; nt < 4; ++nt) {
        v2f b = *(const v2f*)(Kb + (nt * 16 + ln) * KSTR + kk * 4 + 2 * hi);
        S[nt] = wmma4(a, b, S[nt]);
      }
    }

    // ---- online softmax (exp2 domain) ----
    float alpha[8], mnew[8];
#pragma unroll
    for (int v = 0; v < 8; ++v) {
      float t = fmaxf(fmaxf(S[0][v], S[1][v]), fmaxf(S[2][v], S[3][v]));
      t = fmaxf(t, __shfl_xor(t, 1, 32));
      t = fmaxf(t, __shfl_xor(t, 2, 32));
      t = fmaxf(t, __shfl_xor(t, 4, 32));
      t = fmaxf(t, __shfl_xor(t, 8, 32));     // row max over 16 lanes (N dim)
      float mn = fmaxf(mrow[v], t * SCL);
      alpha[v] = __builtin_amdgcn_exp2f(mrow[v] - mn);
      mnew[v] = mn;
    }
#pragma unroll
    for (int nt = 0; nt < 4; ++nt)
#pragma unroll
      for (int v = 0; v < 8; ++v)
        S[nt][v] = __builtin_amdgcn_exp2f(fmaf(S[nt][v], SCL, -mnew[v]));
#pragma unroll
    for (int v = 0; v < 8; ++v) {
      float r = (S[0][v] + S[1][v]) + (S[2][v] + S[3][v]);
      r += __shfl_xor(r, 1, 32);
      r += __shfl_xor(r, 2, 32);
      r += __shfl_xor(r, 4, 32);
      r += __shfl_xor(r, 8, 32);
      lrow[v] = alpha[v] * lrow[v] + r;
      mrow[v] = mnew[v];
    }
#pragma unroll
    for (int ot = 0; ot < 8; ++ot)
#pragma unroll
      for (int v = 0; v < 8; ++v) Oacc[ot][v] *= alpha[v];

    // ---- stage P (C layout -> LDS), then O += P * V ----
#pragma unroll
    for (int nt = 0; nt < 4; ++nt)
#pragma unroll
      for (int v = 0; v < 8; ++v)
        Pw[(v + 8 * hi) * PSTR + nt * 16 + ln] = S[nt][v];
    // same-wave LDS ops are in-order: loads below see the stores above

    const float* Pa = Pw + ln * PSTR + 2 * hi;
#pragma unroll
    for (int kk = 0; kk < 16; ++kk) {
      v2f a = *(const v2f*)(Pa + kk * 4);     // A: P[m][k..k+1]
#pragma unroll
      for (int ot = 0; ot < 8; ++ot) {
        v2f b = *(const v2f*)(Vb + (ot * 16 + ln) * VSTR + kk * 4 + 2 * hi);
        Oacc[ot] = wmma4(a, b, Oacc[ot]);
      }
    }
    __syncthreads();                          // buffer j consumed by everyone
  }

  // ---- epilogue: O / l, coalesced store via LDS staging ----
  float linv[8];
#pragma unroll
  for (int v = 0; v < 8; ++v) linv[v] = 1.0f / lrow[v];

#pragma unroll
  for (int pass = 0; pass < 2; ++pass) {
#pragma unroll
    for (int t4 = 0; t4 < 4; ++t4) {
      int ot = pass * 4 + t4;
#pragma unroll
      for (int v = 0; v < 8; ++v)
        Pw[(v + 8 * hi) * PSTR + t4 * 16 + ln] = Oacc[ot][v] * linv[v];
    }
#pragma unroll
    for (int i = 0; i < 8; ++i) {
      int c = lane + i * 32;
      int r = c >> 4, col = (c & 15) << 2;
      v4f t = *(const v4f*)(Pw + r * PSTR + col);
      *(v4f*)(Oh + (size_t)(q0 + wv * 16 + r) * D_DIM + pass * 64 + col) = t;
    }
  }
}

extern "C" void kernel_launch(void* const* d_in, const int* in_sizes, int n_in,
                              void* d_out, int out_size, void* d_ws, size_t ws_size,
                              hipStream_t stream) {
  const float* q = (const float*)d_in[0];
  const float* k = (const float*)d_in[1];
  const float* v = (const float*)d_in[2];
  float* o = (float*)d_out;
  (void)in_sizes; (void)n_in; (void)out_size; (void)d_ws; (void)ws_size;

  dim3 grid(N_SEQ / BR, 4 * 16);   // (q tiles, B*H)
  dim3 block(THREADS);
  size_t shmem = (size_t)SM_TOTAL * sizeof(float);  // 239616 B (<320KB WGP LDS)
  fa_fwd_f32_wmma<<<grid, block, shmem, stream>>>(q, k, v, o);
}